// FFTLongConv_8710193676345
// MI455X (gfx1250) — compile-verified
//
#include <hip/hip_runtime.h>

typedef __attribute__((ext_vector_type(2))) float v2f;
typedef __attribute__((ext_vector_type(8))) float v8f;

#define WG      256
#define NFFT    16384
#define LHALF   8192
#define ROWLEN  1024
#define NROWS   16

// ---------------------------------------------------------------------------
// Step A: DFT-16 along the stride-1024 dimension as WMMA f32 16x16x4 matmuls.
// src: matrix X[n1][n2] at offset n1*1024+n2 (linear index layout).
// dst: A[k1][n2] * W_16384^{k1*n2} at offset k1*1024+n2.
// ---------------------------------------------------------------------------
__device__ __forceinline__ void dft16_step(const float2* __restrict__ src,
                                           float2* __restrict__ dst,
                                           const v2f wr[4], const v2f wi[4],
                                           const v2f nwi[4],
                                           int lane, int wave)
{
    const int colLane = lane & 15;
    const int koff    = (lane >> 4) << 1;   // K sub-pair selected by lane half
    const int rowBase = (lane >> 4) << 3;   // C/D rows 0..7 or 8..15

    for (int t = wave * 8; t < wave * 8 + 8; ++t) {   // 8 tiles of 16 columns per wave
        const int col = (t << 4) + colLane;
        v8f accR = {};
        v8f accI = {};
#pragma unroll
        for (int k = 0; k < 4; ++k) {
            const int r0 = (k << 2) + koff;           // n1 rows for this B slice
            float2 c0 = src[r0 * ROWLEN + col];
            float2 c1 = src[(r0 + 1) * ROWLEN + col];
            v2f br = {c0.x, c1.x};
            v2f bi = {c0.y, c1.y};
            // Ar += Wr*Xr - Wi*Xi ; Ai += Wr*Xi + Wi*Xr
            accR = __builtin_amdgcn_wmma_f32_16x16x4_f32(false, wr[k],  false, br,
                                                         (short)0, accR, false, false);
            accR = __builtin_amdgcn_wmma_f32_16x16x4_f32(false, nwi[k], false, bi,
                                                         (short)0, accR, false, false);
            accI = __builtin_amdgcn_wmma_f32_16x16x4_f32(false, wr[k],  false, bi,
                                                         (short)0, accI, false, false);
            accI = __builtin_amdgcn_wmma_f32_16x16x4_f32(false, wi[k],  false, br,
                                                         (short)0, accI, false, false);
        }
        // Fuse four-step twiddle W_16384^{k1*n2} into the writeback.
#pragma unroll
        for (int v = 0; v < 8; ++v) {
            const int k1 = rowBase + v;
            float ang = -3.834951969714103e-4f * (float)(k1 * col); // -2pi/16384 * k1*n2
            float s, c;
            __sincosf(ang, &s, &c);
            const float ar = accR[v];
            const float ai = accI[v];
            dst[k1 * ROWLEN + col] = make_float2(ar * c - ai * s, ar * s + ai * c);
        }
    }
}

// ---------------------------------------------------------------------------
// 5 Stockham radix-4 stages: 16 independent 1024-point forward FFTs (rows).
// Twiddles come from the LDS table twd[j] = exp(-2*pi*i*j/1024).
// Data enters in bufA, exits (natural order) in bufA^5 = bufB (odd #stages).
// ---------------------------------------------------------------------------
__device__ __forceinline__ void stockham4_rows(float2* bufA, float2* bufB,
                                               const float2* __restrict__ twd,
                                               int tid)
{
#pragma unroll 1
    for (int st = 0; st < 5; ++st) {
        const float2* src = (st & 1) ? bufB : bufA;
        float2*       dst = (st & 1) ? bufA : bufB;
        const int sh    = st << 1;            // log2(s_i), s_i = 4^st
        const int smask = (1 << sh) - 1;
        const int s_i   = 1 << sh;
        for (int u = tid; u < NROWS * 256; u += WG) {
            const int row  = u >> 8;
            const int j    = u & 255;
            const int q    = j & smask;
            const int sp   = ((j >> sh) << sh);    // s_i * p   (< 256)
            const int base = (row << 10) + q;
            float2 a = src[base + sp];
            float2 b = src[base + sp + 256];
            float2 c = src[base + sp + 512];
            float2 d = src[base + sp + 768];
            float2 w1 = twd[sp];
            float2 w2 = twd[2 * sp];
            float2 w3 = twd[3 * sp];
            float2 apc = make_float2(a.x + c.x, a.y + c.y);
            float2 amc = make_float2(a.x - c.x, a.y - c.y);
            float2 bpd = make_float2(b.x + d.x, b.y + d.y);
            float2 bmd = make_float2(b.x - d.x, b.y - d.y);
            float2 t1 = make_float2(amc.x + bmd.y, amc.y - bmd.x);   // (a-c) - i(b-d)
            float2 t2 = make_float2(apc.x - bpd.x, apc.y - bpd.y);   // (a+c) - (b+d)
            float2 t3 = make_float2(amc.x - bmd.y, amc.y + bmd.x);   // (a-c) + i(b-d)
            const int ob = base + (sp << 2);
            dst[ob]           = make_float2(apc.x + bpd.x, apc.y + bpd.y);
            dst[ob + s_i]     = make_float2(t1.x * w1.x - t1.y * w1.y,
                                            t1.x * w1.y + t1.y * w1.x);
            dst[ob + 2 * s_i] = make_float2(t2.x * w2.x - t2.y * w2.y,
                                            t2.x * w2.y + t2.y * w2.x);
            dst[ob + 3 * s_i] = make_float2(t3.x * w3.x - t3.y * w3.y,
                                            t3.x * w3.y + t3.y * w3.x);
        }
        __syncthreads();
    }
}

// Spectrum position of index k in the [k1][k2] output matrix.
__device__ __forceinline__ int spec_off(int k) {
    return ((k & 15) << 10) + (k >> 4);
}

extern "C" __global__ void __launch_bounds__(WG)
fftconv_kernel(const float* __restrict__ inp, const float* __restrict__ flt,
               const long long* __restrict__ pos, float* __restrict__ out)
{
    extern __shared__ float2 sm[];
    float2* Xb  = sm;                 // 16384 complex
    float2* Yb  = sm + NFFT;          // 16384 complex
    float2* Twd = sm + 2 * NFFT;      // 1024 complex twiddle table (8 KB)

    const int tid  = threadIdx.x;
    const int lane = tid & 31;
    const int wave = tid >> 5;
    const int rowIdx = blockIdx.x;            // 0..2047 = b*256 + d
    const int b = rowIdx >> 8;
    const long long base = (long long)rowIdx * LHALF;

    // Per-lane DFT16 A-operand constants: W16[r][k] = exp(-2*pi*i*r*k/16).
    v2f wr[4], wi[4], nwi[4];
    {
        const int r  = lane & 15;
        const int ko = (lane >> 4) << 1;
#pragma unroll
        for (int k = 0; k < 4; ++k) {
            const int k0 = (k << 2) + ko;
            const float a0 = -0.39269908169872414f * (float)((r * k0) & 15);
            const float a1 = -0.39269908169872414f * (float)((r * (k0 + 1)) & 15);
            float s0, c0, s1, c1;
            __sincosf(a0, &s0, &c0);
            __sincosf(a1, &s1, &c1);
            wr[k]  = (v2f){c0, c1};
            wi[k]  = (v2f){s0, s1};
            nwi[k] = (v2f){-s0, -s1};
        }
    }

    // Twiddle table: Twd[j] = exp(-2*pi*i*j/1024).
    for (int j = tid; j < ROWLEN; j += WG) {
        float s, c;
        __sincosf(-6.135923151542565e-3f * (float)j, &s, &c);
        Twd[j] = make_float2(c, s);
    }

    // Load z = x + i*f with float4 vector loads, zero-pad to 16384 (linear in Xb).
    const float4* inp4 = reinterpret_cast<const float4*>(inp + base);
    const float4* flt4 = reinterpret_cast<const float4*>(flt + base);
    for (int jv = tid; jv < LHALF / 4; jv += WG) {
        const float4 xv = inp4[jv];
        const float4 fv = flt4[jv];
        const int j = jv << 2;
        Xb[j + 0] = make_float2(xv.x, fv.x);
        Xb[j + 1] = make_float2(xv.y, fv.y);
        Xb[j + 2] = make_float2(xv.z, fv.z);
        Xb[j + 3] = make_float2(xv.w, fv.w);
        Xb[j + LHALF + 0] = make_float2(0.0f, 0.0f);
        Xb[j + LHALF + 1] = make_float2(0.0f, 0.0f);
        Xb[j + LHALF + 2] = make_float2(0.0f, 0.0f);
        Xb[j + LHALF + 3] = make_float2(0.0f, 0.0f);
    }
    __syncthreads();

    // Forward FFT: Z = FFT_16384(x + i*f).
    dft16_step(Xb, Yb, wr, wi, nwi, lane, wave);
    __syncthreads();
    stockham4_rows(Yb, Xb, Twd, tid);         // 5 stages -> spectrum in Xb

    // Hermitian split -> X(k), F(k); product; store conj(P)/n^2 linearly in Yb.
    const float invn2 = 1.0f / (16384.0f * 16384.0f);
#pragma unroll 1
    for (int it = 0; it < NFFT / WG; ++it) {
        const int k  = tid + it * WG;
        const int km = (NFFT - k) & (NFFT - 1);
        const float2 Zk = Xb[spec_off(k)];
        const float2 Zm = Xb[spec_off(km)];
        const float2 Zmc = make_float2(Zm.x, -Zm.y);
        const float2 Xf = make_float2(0.5f * (Zk.x + Zmc.x), 0.5f * (Zk.y + Zmc.y));
        const float2 w  = make_float2(Zk.x - Zmc.x, Zk.y - Zmc.y);
        const float2 Ff = make_float2(0.5f * w.y, -0.5f * w.x);
        const float pr = Xf.x * Ff.x - Xf.y * Ff.y;
        const float pi = Xf.x * Ff.y + Xf.y * Ff.x;
        Yb[k] = make_float2(pr * invn2, -pi * invn2);   // conj, pre-scaled
    }
    __syncthreads();

    // Inverse FFT via forward machinery on the conjugated spectrum.
    dft16_step(Yb, Xb, wr, wi, nwi, lane, wave);
    __syncthreads();
    stockham4_rows(Xb, Yb, Twd, tid);         // 5 stages -> result in Yb

    // Masked, vectorized writeback of first L samples: y[t] = Re(Yb[spec_off(t)]).
    const long long pbase = (long long)b * LHALF;
    float4* out4 = reinterpret_cast<float4*>(out + base);
    for (int tb = tid; tb < LHALF / 4; tb += WG) {
        const int t0 = tb << 2;
        float4 r;
        r.x = (pos[pbase + t0 + 0] != -1LL) ? Yb[spec_off(t0 + 0)].x : 0.0f;
        r.y = (pos[pbase + t0 + 1] != -1LL) ? Yb[spec_off(t0 + 1)].x : 0.0f;
        r.z = (pos[pbase + t0 + 2] != -1LL) ? Yb[spec_off(t0 + 2)].x : 0.0f;
        r.w = (pos[pbase + t0 + 3] != -1LL) ? Yb[spec_off(t0 + 3)].x : 0.0f;
        out4[tb] = r;
    }
}

extern "C" void kernel_launch(void* const* d_in, const int* in_sizes, int n_in,
                              void* d_out, int out_size, void* d_ws, size_t ws_size,
                              hipStream_t stream) {
    const float*     inp = (const float*)d_in[0];
    const float*     flt = (const float*)d_in[1];
    const long long* pos = (const long long*)d_in[2];
    float* out = (float*)d_out;

    const int nRows = 8 * 256;                               // B * D
    const size_t lds = (size_t)(2 * NFFT + ROWLEN) * sizeof(float2);  // 264 KB
    hipLaunchKernelGGL(fftconv_kernel, dim3(nRows), dim3(WG), lds, stream,
                       inp, flt, pos, out);
}